// TopKFaissSoftmax_14267881357590
// MI455X (gfx1250) — compile-verified
//
#include <hip/hip_runtime.h>

#define TOKENS 4096
#define DMODEL 1024
#define VOCAB  50257
#define KSEL   64
#define CHUNK  1024
#define NCHUNK 50                 // ceil(50257/1024)
#define RTILE  32                 // rows per block (2 WMMA sub-tiles)
#define DSTRIDE 1026              // padded LDS row stride: 8*1026 % 64 == 16
                                  // -> half-waves hit disjoint bank sets

typedef __attribute__((ext_vector_type(2))) float v2f;
typedef __attribute__((ext_vector_type(8))) float v8f;

// ---------------------------------------------------------------------------
// Kernel 0: squared row norms of x and b. One wave per row.
// ---------------------------------------------------------------------------
__global__ __launch_bounds__(256) void knn_norms_kernel(
    const float* __restrict__ x, const float* __restrict__ b,
    float* __restrict__ xsq, float* __restrict__ bsq) {
  int row  = blockIdx.x * (blockDim.x >> 5) + (threadIdx.x >> 5);
  int lane = threadIdx.x & 31;
  const float* src;
  float* dst;
  int r;
  if (row < TOKENS) {
    src = x + (size_t)row * DMODEL; dst = xsq; r = row;
  } else {
    r = row - TOKENS;
    if (r >= VOCAB) return;
    src = b + (size_t)r * DMODEL; dst = bsq;
  }
  float s = 0.0f;
  for (int k = lane; k < DMODEL; k += 32) {
    float v = src[k];
    s = fmaf(v, v, s);
  }
  #pragma unroll
  for (int off = 16; off; off >>= 1) s += __shfl_xor(s, off, 32);
  if (lane == 0) dst[r] = s;
}

// ---------------------------------------------------------------------------
// Kernel 1: f32 WMMA GEMM for a 32-row x 1024-col chunk (2 row sub-tiles per
// wave share each B fragment), distances into LDS (bank-conflict-free padded
// stride), then exact per-row top-64 via in-register bitonic sort + 32-way
// tournament merge. Grid: (rowTile = TOKENS/32, chunk = 50); 256 thr = 8 waves.
// ---------------------------------------------------------------------------
__global__ __launch_bounds__(256) void knn_gemm_select_kernel(
    const float* __restrict__ x, const float* __restrict__ b,
    const float* __restrict__ xsq, const float* __restrict__ bsq,
    float* __restrict__ cand) {
  __shared__ float distbuf[RTILE * DSTRIDE];   // ~128.2 KB

  const int rowTile  = blockIdx.x;             // 0..127
  const int chunk    = blockIdx.y;             // 0..49
  const int wave     = threadIdx.x >> 5;       // 0..7
  const int lane     = threadIdx.x & 31;
  const int row0     = rowTile * RTILE;
  const int colBase  = chunk * CHUNK;

  const int m       = lane & 15;               // A row / B column within tile
  const int halfsel = lane >> 4;               // 0 or 1
  const int k0      = halfsel * 2;             // this lane's K pair offset

  const float* xrow0 = x + (size_t)(row0 + m) * DMODEL;
  const float* xrow1 = x + (size_t)(row0 + 16 + m) * DMODEL;

  // Hoist the block-invariant x-norms this lane will need (C-layout rows).
  float xq0[8], xq1[8];
  #pragma unroll
  for (int i = 0; i < 8; ++i) {
    xq0[i] = xsq[row0 + i + 8 * halfsel];
    xq1[i] = xsq[row0 + 16 + i + 8 * halfsel];
  }

  // ---- GEMM: 8 column tiles of 16 per wave; 2 row sub-tiles per B frag ----
  for (int t = 0; t < 8; ++t) {
    const int tileCol0 = (wave * 8 + t) * 16;
    const int coln = colBase + tileCol0 + m;               // this lane's b row
    const int colc = (coln < VOCAB) ? coln : (VOCAB - 1);  // safe clamp
    const float* brow = b + (size_t)colc * DMODEL;

    // b-norm in flight during the whole k-loop
    const float bn = bsq[colc];

    // Prefetch next tile's b row (stream ahead of the k-loop).
    if (t < 7) {
      int nColn = colBase + tileCol0 + 16 + m;
      int nColc = (nColn < VOCAB) ? nColn : (VOCAB - 1);
      const char* np = (const char*)(b + (size_t)nColc * DMODEL);
      __builtin_prefetch(np, 0, 1);
      __builtin_prefetch(np + 2048, 0, 1);
    }

    v8f acc0 = {0.f, 0.f, 0.f, 0.f, 0.f, 0.f, 0.f, 0.f};
    v8f acc1 = {0.f, 0.f, 0.f, 0.f, 0.f, 0.f, 0.f, 0.f};
    for (int k = 0; k < DMODEL; k += 4) {
      v2f a0 = *(const v2f*)(xrow0 + k + k0);
      v2f a1 = *(const v2f*)(xrow1 + k + k0);
      v2f bv = *(const v2f*)(brow + k + k0);
      acc0 = __builtin_amdgcn_wmma_f32_16x16x4_f32(
          false, a0, false, bv, (short)0, acc0, false, false);
      acc1 = __builtin_amdgcn_wmma_f32_16x16x4_f32(
          false, a1, false, bv, (short)0, acc1, false, false);
    }

    // dist[r][c] = ||x_r||^2 + ||b_c||^2 - 2 * (x_r . b_c)
    const bool colValid = (coln < VOCAB);
    const int ldsCol = tileCol0 + m;
    #pragma unroll
    for (int i = 0; i < 8; ++i) {
      const int r = i + 8 * halfsel;           // C layout: VGPR i -> row i+8*(lane>>4)
      float d0 = xq0[i] + bn - 2.0f * acc0[i];
      float d1 = xq1[i] + bn - 2.0f * acc1[i];
      distbuf[r * DSTRIDE + ldsCol]        = colValid ? d0 : __builtin_inff();
      distbuf[(16 + r) * DSTRIDE + ldsCol] = colValid ? d1 : __builtin_inff();
    }
  }
  __syncthreads();

  // ---- per-row top-64 of this 1024-col chunk: wave w owns rows 4w..4w+3 ---
  for (int rr = 0; rr < 4; ++rr) {
    const int r = wave * 4 + rr;
    float* rowbuf = distbuf + r * DSTRIDE;
    const size_t outBase = ((size_t)(row0 + r) * NCHUNK + chunk) * KSEL;

    // 1) each lane grabs its interleaved 32-element slice (conflict-free)
    float v[32];
    #pragma unroll
    for (int j = 0; j < 32; ++j) v[j] = rowbuf[j * 32 + lane];

    // 2) in-register bitonic sort (ascending), constant indices -> pure VALU
    #pragma unroll
    for (int kk = 2; kk <= 32; kk <<= 1) {
      #pragma unroll
      for (int j = kk >> 1; j > 0; j >>= 1) {
        #pragma unroll
        for (int i = 0; i < 32; ++i) {
          int l = i ^ j;
          if (l > i) {
            bool up = ((i & kk) == 0);
            float lo = fminf(v[i], v[l]);
            float hi = fmaxf(v[i], v[l]);
            v[i] = up ? lo : hi;
            v[l] = up ? hi : lo;
          }
        }
      }
    }

    // 3) write sorted slices back so they can be popped with a dynamic index
    #pragma unroll
    for (int j = 0; j < 32; ++j) rowbuf[j * 32 + lane] = v[j];

    // 4) 32-way tournament merge: pop the 64 smallest, ascending
    int   ptr  = 0;
    float head = v[0];
    for (int t = 0; t < KSEL; ++t) {
      float bvv = head;
      int   bl  = lane;
      #pragma unroll
      for (int off = 16; off; off >>= 1) {
        float ov = __shfl_xor(bvv, off, 32);
        int   ol = __shfl_xor(bl, off, 32);
        if (ov < bvv || (ov == bvv && ol < bl)) { bvv = ov; bl = ol; }
      }
      if (lane == 0) cand[outBase + t] = bvv;
      if (lane == bl) {
        ++ptr;
        head = (ptr < 32) ? rowbuf[ptr * 32 + lane] : __builtin_inff();
      }
    }
  }
}

// ---------------------------------------------------------------------------
// Kernel 2: merge the 50 sorted 64-runs per row -> 64 smallest, ascending.
// One wave per row; lane l owns runs 2l and 2l+1 (50 runs over 25 lanes).
// ---------------------------------------------------------------------------
__global__ __launch_bounds__(256) void knn_merge_kernel(
    const float* __restrict__ cand, float* __restrict__ out) {
  const int wave = threadIdx.x >> 5;
  const int lane = threadIdx.x & 31;
  const int row  = blockIdx.x * 8 + wave;
  if (row >= TOKENS) return;

  const size_t base = (size_t)row * NCHUNK * KSEL;
  const int r0 = 2 * lane, r1 = 2 * lane + 1;
  int   p0 = 0, p1 = 0;
  float h0 = (r0 < NCHUNK) ? cand[base + (size_t)r0 * KSEL] : __builtin_inff();
  float h1 = (r1 < NCHUNK) ? cand[base + (size_t)r1 * KSEL] : __builtin_inff();

  for (int t = 0; t < KSEL; ++t) {
    float head = fminf(h0, h1);
    float bvv = head;
    int   bl  = lane;
    #pragma unroll
    for (int off = 16; off; off >>= 1) {
      float ov = __shfl_xor(bvv, off, 32);
      int   ol = __shfl_xor(bl, off, 32);
      if (ov < bvv || (ov == bvv && ol < bl)) { bvv = ov; bl = ol; }
    }
    if (lane == 0) out[(size_t)row * KSEL + t] = bvv;
    if (lane == bl) {
      if (h0 <= h1) {
        ++p0;
        h0 = (p0 < KSEL) ? cand[base + (size_t)r0 * KSEL + p0] : __builtin_inff();
      } else {
        ++p1;
        h1 = (p1 < KSEL) ? cand[base + (size_t)r1 * KSEL + p1] : __builtin_inff();
      }
    }
  }
}

// ---------------------------------------------------------------------------
// Host launcher
// ---------------------------------------------------------------------------
extern "C" void kernel_launch(void* const* d_in, const int* in_sizes, int n_in,
                              void* d_out, int out_size, void* d_ws, size_t ws_size,
                              hipStream_t stream) {
  const float* x = (const float*)d_in[0];      // [4096,1024] fp32
  const float* b = (const float*)d_in[1];      // [50257,1024] fp32
  // d_in[2] (target) and d_in[3] (k) unused: k is compile-time 64.
  float* out = (float*)d_out;                  // [4096,64] fp32

  // Workspace layout (floats): xsq | bsq (padded) | candidates (52.4 MB)
  float* xsq  = (float*)d_ws;                          // 4096
  float* bsq  = xsq + TOKENS;                          // 50257 -> pad to 50432
  float* cand = bsq + ((VOCAB + 255) & ~255);          // 4096*50*64 floats

  // Kernel 0: norms (8 waves/block, one row per wave)
  {
    int rows = TOKENS + VOCAB;
    int blocks = (rows + 7) / 8;
    knn_norms_kernel<<<blocks, 256, 0, stream>>>(x, b, xsq, bsq);
  }

  // Kernel 1: WMMA GEMM + per-chunk exact top-64.
  // grid.x = row tiles (fast) so all blocks of a chunk share the b-slice in L2.
  {
    dim3 grid(TOKENS / RTILE, NCHUNK);
    knn_gemm_select_kernel<<<grid, 256, 0, stream>>>(x, b, xsq, bsq, cand);
  }

  // Kernel 2: 50-way sorted-run merge per row, ascending output.
  knn_merge_kernel<<<(TOKENS + 7) / 8, 256, 0, stream>>>(cand, out);
}